// NLRWDense_55757265436993
// MI455X (gfx1250) — compile-verified
//
#include <hip/hip_runtime.h>
#include <math.h>

// ---------------------------------------------------------------------------
// NLRWDense on MI455X (gfx1250), full-f32 pipeline on the WMMA matrix cores.
//
//   na = ||x_i||^2, nb = ||k_j||^2
//   Tul = exp(DP*sqrt(max(na_i - 2 X K^T + nb_j, 0)))      [B,L]
//   Tuu = exp(DP*sqrt(max(2 na_i - 2 X X^T, 0)))           [B,B]  (faithful 2*na_i)
//   S_i = rowsum(Tul) + rowsum(Tuu);  Pul = Tul/S, Puu = Tuu/S
//   out = inv(I - Puu) Pul  via Neumann iteration X <- Pul + rS*(Tuu X)
//         (||Puu||_inf < 1 since rows of [Pul|Puu] sum to 1, entries > 0)
//
// Tuu is kept UNSCALED; the row scale rS[i] is folded into the iteration
// epilogue, saving a 128 MB read+write normalization pass over Puu.
// ---------------------------------------------------------------------------

typedef __attribute__((ext_vector_type(2))) float v2f;
typedef __attribute__((ext_vector_type(8))) float v8f;

// Explicit global-address-space pointers: guarantees global_load/global_store
// (LOADcnt/STOREcnt only) instead of flat_* (which also burns DScnt and the
// LDS data path). Global VA == generic VA for global memory on AMDGCN, so an
// inttoptr round-trip is a correct address-space assertion.
#define GAS __attribute__((address_space(1)))
typedef GAS const float* gcf;
typedef GAS float*       gf;
typedef GAS const v2f*   gcv2;
#define TO_GCF(p) ((gcf)(unsigned long long)(p))
#define TO_GF(p)  ((gf)(unsigned long long)(p))

#define DP 0.05f

// ---------------------------------------------------------------------------
// Row squared-norm: one block per row.
__global__ __launch_bounds__(256)
void row_norms_kernel(const float* __restrict__ X, float* __restrict__ out, int cols) {
    const int row = blockIdx.x;
    const float* r = X + (size_t)row * cols;
    float s = 0.f;
    for (int c = threadIdx.x; c < cols; c += blockDim.x) {
        float v = r[c];
        s += v * v;
    }
    __shared__ float red[256];
    red[threadIdx.x] = s;
    __syncthreads();
    for (int off = 128; off > 0; off >>= 1) {
        if ((int)threadIdx.x < off) red[threadIdx.x] += red[threadIdx.x + off];
        __syncthreads();
    }
    if (threadIdx.x == 0) out[row] = red[0];
}

// ---------------------------------------------------------------------------
// S_i = sum_j Tul[i,j] + sum_j Tuu[i,j];  store 1/S_i. One block per row.
__global__ __launch_bounds__(256)
void row_sums_recip_kernel(const float* __restrict__ Tul, const float* __restrict__ Tuu,
                           float* __restrict__ rS, int L, int Bn) {
    const int row = blockIdx.x;
    const float* rul = Tul + (size_t)row * L;
    const float* ruu = Tuu + (size_t)row * Bn;
    float s = 0.f;
    for (int c = threadIdx.x; c < L;  c += blockDim.x) s += rul[c];
    for (int c = threadIdx.x; c < Bn; c += blockDim.x) s += ruu[c];
    __shared__ float red[256];
    red[threadIdx.x] = s;
    __syncthreads();
    for (int off = 128; off > 0; off >>= 1) {
        if ((int)threadIdx.x < off) red[threadIdx.x] += red[threadIdx.x + off];
        __syncthreads();
    }
    if (threadIdx.x == 0) rS[row] = 1.0f / red[0];
}

// ---------------------------------------------------------------------------
// T[row, col] *= rS[row]   (used only for Tul -> Pul, 16 MB)
__global__ __launch_bounds__(256)
void scale_rows_kernel(float* __restrict__ T, const float* __restrict__ rS, int cols) {
    const int col = blockIdx.x * blockDim.x + threadIdx.x;
    const int row = blockIdx.y;
    if (col < cols) T[(size_t)row * cols + col] *= rS[row];
}

// ---------------------------------------------------------------------------
// f32 WMMA GEMM computing G = A (x) B, fused epilogue.
//   BT = true :  G[i,j] = sum_k A[i,k] * B[j,k]   (B row-major [N,K], i.e. A*B^T)
//   BT = false:  G[i,j] = sum_k A[i,k] * B[k,j]   (B row-major [K,N])
// MODE 0: C = exp(DP*sqrt(max(na[i] - 2G + nb[j], 0)))
// MODE 1: C = exp(DP*sqrt(max(2*na[i] - 2G      , 0)))
// MODE 2: C = addm[i,j] + rS[i]*G
//
// Block: 256 threads = 8 waves; block tile 128x256; wave tile 64x64
// (4x4 subtiles of 16x16 -> 128 accumulator VGPRs, 16 WMMA per 8 frag loads).
// Requires M % 128 == 0, N % 256 == 0, K % 4 == 0.
//
// Loads use per-fragment global(AS1) pointer streams advanced in the K loop:
// the backend folds the unrolled advance into IOFFSET immediates and emits one
// pointer bump per stream per unrolled chunk.
//
// Fragment layouts per CDNA5 ISA 7.12.2 (wave32):
//   A 16x4 f32 : 2 VGPRs; lanes 0-15 hold K={0,1}, lanes 16-31 hold K={2,3}
//   B 4x16 f32 : 2 VGPRs; mirrored (lane = N, lane halves split K)
//   C/D 16x16  : 8 VGPRs; row = v + 8*(lane>>4), col = lane&15
template<int MODE, bool BT>
__global__ __launch_bounds__(256)
void wmma_gemm_kernel(const float* __restrict__ A, const float* __restrict__ B,
                      float* __restrict__ C,
                      const float* __restrict__ na, const float* __restrict__ nb,
                      const float* __restrict__ addm, const float* __restrict__ rS,
                      int N, int K) {
    const int lane = threadIdx.x & 31;
    const int w    = threadIdx.x >> 5;
    const int m16  = lane & 15;       // row/col within a 16x16 tile
    const int p    = lane >> 4;       // 0/1: which K/M half this lane owns
    const int row0 = blockIdx.y * 128 + (w & 1) * 64;   // 2 wave rows
    const int col0 = blockIdx.x * 256 + (w >> 1) * 64;  // 4 wave cols

    // Per-fragment global pointer streams.
    gcf pa[4];
    gcf pb[4];
#pragma unroll
    for (int mt = 0; mt < 4; ++mt)
        pa[mt] = TO_GCF(A + (size_t)(row0 + mt * 16 + m16) * K + 2 * p);
#pragma unroll
    for (int nt = 0; nt < 4; ++nt) {
        if (BT) pb[nt] = TO_GCF(B + (size_t)(col0 + nt * 16 + m16) * K + 2 * p);
        else    pb[nt] = TO_GCF(B + (size_t)(2 * p) * N + (col0 + nt * 16 + m16));
    }

    v8f acc[4][4] = {};

    for (int k0 = 0; k0 < K; k0 += 4) {
        v2f a[4], b[4];
#pragma unroll
        for (int mt = 0; mt < 4; ++mt) {
            a[mt] = *(gcv2)pa[mt];
            pa[mt] += 4;
        }
#pragma unroll
        for (int nt = 0; nt < 4; ++nt) {
            if (BT) {
                b[nt] = *(gcv2)pb[nt];
                pb[nt] += 4;
            } else {
                v2f t;
                t.x = pb[nt][0];
                t.y = pb[nt][N];
                b[nt] = t;
                pb[nt] += (size_t)4 * N;
            }
        }
#pragma unroll
        for (int mt = 0; mt < 4; ++mt)
#pragma unroll
            for (int nt = 0; nt < 4; ++nt)
                acc[mt][nt] = __builtin_amdgcn_wmma_f32_16x16x4_f32(
                    false, a[mt], false, b[nt], (short)0, acc[mt][nt], false, false);
    }

    // Fused epilogue + store (full tiles, no bounds checks needed).
    const gcf gna = TO_GCF(na);
    const gcf gnb = TO_GCF(nb);
    const gcf gad = TO_GCF(addm);
    const gcf grs = TO_GCF(rS);
    const gf  gc  = TO_GF(C);
#pragma unroll
    for (int mt = 0; mt < 4; ++mt) {
#pragma unroll
        for (int nt = 0; nt < 4; ++nt) {
#pragma unroll
            for (int v = 0; v < 8; ++v) {
                const int row = row0 + mt * 16 + v + 8 * p;
                const int col = col0 + nt * 16 + m16;
                const size_t coff = (size_t)row * N + col;
                const float g = acc[mt][nt][v];
                float outv;
                if (MODE == 0) {
                    float t = gna[row] - 2.0f * g + gnb[col];
                    t = fmaxf(t, 0.0f);
                    outv = __expf(DP * sqrtf(t));
                } else if (MODE == 1) {
                    float t = 2.0f * gna[row] - 2.0f * g;
                    t = fmaxf(t, 0.0f);
                    outv = __expf(DP * sqrtf(t));
                } else {
                    outv = fmaf(grs[row], g, gad[coff]);
                }
                gc[coff] = outv;
            }
        }
    }
}

// ---------------------------------------------------------------------------
extern "C" void kernel_launch(void* const* d_in, const int* in_sizes, int n_in,
                              void* d_out, int out_size, void* d_ws, size_t ws_size,
                              hipStream_t stream) {
    const int D  = 512;
    const int Bn = in_sizes[0] / D;   // 4096
    const int L  = in_sizes[1] / D;   // 1024

    const float* Xin = (const float*)d_in[0];
    const float* Krn = (const float*)d_in[1];
    float* out = (float*)d_out;

    // Workspace layout (floats): na[Bn] nb[L] rS[Bn] Tul[Bn*L] Tuu[Bn*Bn] X0[Bn*L]
    float* ws  = (float*)d_ws;
    float* na  = ws;
    float* nb  = na + Bn;
    float* rS  = nb + L;
    float* Tul = rS + Bn;
    float* Tuu = Tul + (size_t)Bn * L;
    float* X0  = Tuu + (size_t)Bn * Bn;

    const dim3 blk(256);

    // 1) squared row norms
    row_norms_kernel<<<Bn, blk, 0, stream>>>(Xin, na, D);
    row_norms_kernel<<<L,  blk, 0, stream>>>(Krn, nb, D);

    // 2) Tul = exp(DP*sqrt(max(na - 2 X K^T + nb, 0)))     grid (L/256, Bn/128)
    wmma_gemm_kernel<0, true><<<dim3(L / 256, Bn / 128), blk, 0, stream>>>(
        Xin, Krn, Tul, na, nb, nullptr, nullptr, L, D);

    // 3) Tuu = exp(DP*sqrt(max(2 na - 2 X X^T, 0)))        grid (Bn/256, Bn/128)
    wmma_gemm_kernel<1, true><<<dim3(Bn / 256, Bn / 128), blk, 0, stream>>>(
        Xin, Xin, Tuu, na, nullptr, nullptr, nullptr, Bn, D);

    // 4) row sums -> 1/S; normalize Tul -> Pul in place (Tuu stays unscaled,
    //    rS is applied inside the iteration epilogue).
    row_sums_recip_kernel<<<Bn, blk, 0, stream>>>(Tul, Tuu, rS, L, Bn);
    scale_rows_kernel<<<dim3(L / 256, Bn), blk, 0, stream>>>(Tul, rS, L);

    // 5) Neumann iteration  X <- Pul + rS*(Tuu X), X_0 = Pul.
    //    ||Puu||_inf < 1; 128 steps contract well below f32 epsilon.
    //    Ping-pong X0 <-> d_out; NITER even so the final GEMM writes d_out.
    const int NITER = 128;
    for (int i = 0; i < NITER; ++i) {
        const float* xin = (i == 0) ? Tul : ((i & 1) ? X0 : out);
        float* xout = (i & 1) ? out : X0;
        wmma_gemm_kernel<2, false><<<dim3(L / 256, Bn / 128), blk, 0, stream>>>(
            Tuu, xin, xout, nullptr, nullptr, Tul, rS, L, Bn);
    }
}